// AVWGCN_88338887344443
// MI455X (gfx1250) — compile-verified
//
#include <hip/hip_runtime.h>

// Problem constants (match reference)
#define B_   32
#define N_   2048
#define C_   64
#define D_   8
#define G_   4
#define NPC_ 512
#define TW_  73   // 1 + 8 + 64 combined TE width
#define TWP_ 80   // padded to 5 WMMA row tiles

typedef float v2f __attribute__((ext_vector_type(2)));
typedef float v8f __attribute__((ext_vector_type(8)));

// ws layout (float offsets)
#define OFF_T    0
#define OFF_TSC  (OFF_T   + N_*TWP_)        // TscT stored transposed [TWP_][N_]
#define OFF_CL   (OFF_TSC + TWP_*N_)        // [G][B][TWP][C]
#define OFF_EH   (OFF_CL  + G_*B_*TWP_*C_)  // [B][TWP][C]
#define OFF_Z    (OFF_EH  + B_*TWP_*C_)     // [B][N][C]
#define OFF_CTR  (OFF_Z   + (size_t)B_*N_*C_)

#define WMMA_F32(a, b, c) \
  __builtin_amdgcn_wmma_f32_16x16x4_f32(false, (a), false, (b), (short)0, (c), false, false)

// ---------------------------------------------------------------------------
// Kernel 0: build combined Kronecker basis T[n][e'] and poly-scaled transpose
//           TscT[e'][n]; zero nonzero counters.
// ---------------------------------------------------------------------------
__global__ void k_build(const float* __restrict__ E, const float* __restrict__ poly,
                        float* __restrict__ T, float* __restrict__ TscT,
                        unsigned int* __restrict__ ctr) {
  const int n = blockIdx.x * blockDim.x + threadIdx.x;
  if (n == 0) { ctr[0] = 0u; ctr[1] = 0u; }
  if (n >= N_) return;
  const int g = n / NPC_;
  float e[D_];
#pragma unroll
  for (int d = 0; d < D_; ++d) e[d] = E[n * D_ + d];
  const float p0 = poly[g * 3 + 0];
  const float p1 = poly[g * 3 + 1];
  const float p2 = poly[g * 3 + 2];

  T[n * TWP_ + 0] = 1.0f;
  TscT[0 * N_ + n] = p0;
#pragma unroll
  for (int d = 0; d < D_; ++d) {
    T[n * TWP_ + 1 + d] = e[d];
    TscT[(1 + d) * N_ + n] = p1 * e[d];
  }
#pragma unroll
  for (int i = 0; i < D_; ++i)
#pragma unroll
    for (int j = 0; j < D_; ++j) {
      const int idx = 9 + i * D_ + j;
      const float v = e[i] * e[j];
      T[n * TWP_ + idx] = v;
      TscT[idx * N_ + n] = p2 * v;
    }
#pragma unroll
  for (int idx = TW_; idx < TWP_; ++idx) {   // zero pad rows
    T[n * TWP_ + idx] = 0.0f;
    TscT[idx * N_ + n] = 0.0f;
  }
}

// ---------------------------------------------------------------------------
// Kernel 1: per-client partial aggregates  CL[g,b] = T_g^T (80x512) @ H (512x64)
// One block per (g,b); 20 waves, each owns one 16x16 output tile. WMMA f32 K=4.
// Also counts nonzeros of CL for the `sp` statistic.
// ---------------------------------------------------------------------------
__global__ void __launch_bounds__(640)
k_partial(const float* __restrict__ x, const float* __restrict__ mask,
          const float* __restrict__ T, float* __restrict__ CL,
          unsigned int* __restrict__ ctr) {
  const int g = blockIdx.x >> 5;
  const int b = blockIdx.x & 31;
  const int w = threadIdx.x >> 5;
  const int lane = threadIdx.x & 31;
  const int mt = w >> 2;          // e' tile 0..4
  const int ct = w & 3;           // c  tile 0..3
  const int lm = lane & 15;
  const int kb = (lane >> 4) << 1;       // 0 or 2 (K split across lane halves)
  const int nbase = g * NPC_;
  const int erow = mt * 16 + lm;
  const int ccol = ct * 16 + lm;

  v8f acc = {0.f, 0.f, 0.f, 0.f, 0.f, 0.f, 0.f, 0.f};
  for (int kk = 0; kk < NPC_; kk += 4) {
    const int k0 = nbase + kk + kb;     // node index for this lane-half
    v2f a, bb;
    a.x = T[(k0    ) * TWP_ + erow];    // A[m=e',k=node] = T[node][e']
    a.y = T[(k0 + 1) * TWP_ + erow];
    const int xi0 = (b * N_ + k0) * C_ + ccol;
    bb.x = x[xi0] * mask[xi0];          // B[k=node,n=c] = H[b,node,c]
    bb.y = x[xi0 + C_] * mask[xi0 + C_];
    acc = WMMA_F32(a, bb, acc);
  }

  const int ebase = mt * 16 + ((lane >> 4) << 3);  // C/D rows split at M=8
  unsigned int cnt = 0;
#pragma unroll
  for (int v = 0; v < 8; ++v) {
    const float val = acc[v];
    CL[(((g * B_ + b) * TWP_) + ebase + v) * C_ + ccol] = val;
    cnt += (val != 0.0f) ? 1u : 0u;     // pad rows are exact zeros -> contribute 0
  }
  atomicAdd(&ctr[0], cnt);
}

// ---------------------------------------------------------------------------
// Kernel 2: EH = sum over clients of CL; count nonzeros of EH (tran statistic)
// ---------------------------------------------------------------------------
__global__ void k_reduce(const float* __restrict__ CL, float* __restrict__ EH,
                         unsigned int* __restrict__ ctr) {
  const int idx = blockIdx.x * blockDim.x + threadIdx.x;  // < B*TWP*C
  float s = 0.f;
#pragma unroll
  for (int g = 0; g < G_; ++g) s += CL[g * (B_ * TWP_ * C_) + idx];
  EH[idx] = s;
  const unsigned long long m = __ballot(s != 0.0f);
  if ((threadIdx.x & 31) == 0) atomicAdd(&ctr[1], (unsigned int)__popcll(m));
}

// ---------------------------------------------------------------------------
// Kernel 3: Z = H + Tsc (2048x80) @ EH[b] (80x64).  8 waves/block; each wave
// computes a 16x64 strip (A fragment reused across 4 c-tiles).
// ---------------------------------------------------------------------------
__global__ void __launch_bounds__(256)
k_zcombine(const float* __restrict__ x, const float* __restrict__ mask,
           const float* __restrict__ TscT, const float* __restrict__ EH,
           float* __restrict__ Z) {
  const int wg = blockIdx.x * 8 + (threadIdx.x >> 5);
  const int lane = threadIdx.x & 31;
  const int b = wg >> 7;                // /128 node-tiles
  const int n0 = (wg & 127) * 16;
  const int lm = lane & 15;
  const int kb = (lane >> 4) << 1;
  const float* __restrict__ EHb = EH + b * TWP_ * C_;

  v8f acc[4];
#pragma unroll
  for (int t = 0; t < 4; ++t) acc[t] = (v8f){0.f,0.f,0.f,0.f,0.f,0.f,0.f,0.f};

  for (int k0 = 0; k0 < TWP_; k0 += 4) {
    const int k = k0 + kb;
    v2f a;
    a.x = TscT[(k    ) * N_ + n0 + lm];  // A[m=node,k=e'] (coalesced across lanes)
    a.y = TscT[(k + 1) * N_ + n0 + lm];
#pragma unroll
    for (int ct = 0; ct < 4; ++ct) {
      v2f bb;
      bb.x = EHb[(k    ) * C_ + ct * 16 + lm];
      bb.y = EHb[(k + 1) * C_ + ct * 16 + lm];
      acc[ct] = WMMA_F32(a, bb, acc[ct]);
    }
  }

  const int rbase = (lane >> 4) * 8;
#pragma unroll
  for (int v = 0; v < 8; ++v) {
    const int n = n0 + rbase + v;
#pragma unroll
    for (int ct = 0; ct < 4; ++ct) {
      const int gi = (b * N_ + n) * C_ + ct * 16 + lm;
      Z[gi] = x[gi] * mask[gi] + acc[ct][v];   // residual H + Zp
    }
  }
}

// ---------------------------------------------------------------------------
// Kernel 4: per-node adaptive MLP. One block per node: build W_n = E[n]·wp[g]
// and bias_n in LDS, stage Z[:,n,:], then 32x64x64 GEMM via 8 WMMA tiles.
// ---------------------------------------------------------------------------
__global__ void __launch_bounds__(256)
k_out(const float* __restrict__ E, const float* __restrict__ wp,
      const float* __restrict__ bp, const float* __restrict__ Z,
      float* __restrict__ out) {
  const int n = blockIdx.x;
  const int g = n / NPC_;
  const int t = threadIdx.x;

  __shared__ float sW[C_][C_ + 1];   // W_n, padded vs bank conflicts
  __shared__ float sZ[B_][C_ + 1];   // Z[:, n, :]
  __shared__ float sb[C_];

  float e[D_];
#pragma unroll
  for (int d = 0; d < D_; ++d) e[d] = E[n * D_ + d];

  {  // W_n[i][o] = sum_d e[d]*wp[g,d,i,o]   (o consecutive across threads)
    const int o = t & 63;
    const int ibase = (t >> 6) << 4;
    for (int i = ibase; i < ibase + 16; ++i) {
      float acc = 0.f;
#pragma unroll
      for (int d = 0; d < D_; ++d)
        acc += e[d] * wp[((g * D_ + d) * C_ + i) * C_ + o];
      sW[i][o] = acc;
    }
    if (t < C_) {
      float acc = 0.f;
#pragma unroll
      for (int d = 0; d < D_; ++d) acc += e[d] * bp[(g * D_ + d) * C_ + t];
      sb[t] = acc;
    }
  }
#pragma unroll
  for (int r = 0; r < 8; ++r) {      // stage Z rows for this node
    const int idx = t + r * 256;
    sZ[idx >> 6][idx & 63] = Z[((idx >> 6) * N_ + n) * C_ + (idx & 63)];
  }
  __syncthreads();

  const int w = t >> 5, lane = t & 31;
  const int mt = w >> 2, ctile = w & 3;
  const int lm = lane & 15, kb = (lane >> 4) << 1;

  v8f acc = {0.f, 0.f, 0.f, 0.f, 0.f, 0.f, 0.f, 0.f};
#pragma unroll
  for (int k0 = 0; k0 < C_; k0 += 4) {
    const int k = k0 + kb;
    v2f a, bb;
    a.x = sZ[mt * 16 + lm][k];              // A[m=batch,k=c_in]
    a.y = sZ[mt * 16 + lm][k + 1];
    bb.x = sW[k][ctile * 16 + lm];          // B[k=c_in,n=c_out]
    bb.y = sW[k + 1][ctile * 16 + lm];
    acc = WMMA_F32(a, bb, acc);
  }

  const int rbase = (lane >> 4) * 8;
#pragma unroll
  for (int v = 0; v < 8; ++v) {
    const int brow = mt * 16 + rbase + v;
    const int o = ctile * 16 + lm;
    out[(brow * N_ + n) * C_ + o] = acc[v] + sb[o];
  }
}

// ---------------------------------------------------------------------------
// Kernel 5: scalars sp / tran from nonzero counters (denominators constant).
// ---------------------------------------------------------------------------
__global__ void k_final(const unsigned int* __restrict__ ctr,
                        float* __restrict__ out_scalars) {
  const float num  = (float)ctr[0];
  const float num2 = (float)ctr[1];
  const float den  = (float)(G_ * B_ * TW_ * C_);   // 598016
  const float den2 = (float)(B_ * TW_ * C_);        // 149504
  const float sp   = 1.0f - num / (den + 1e-10f);
  const float tran = (1.0f - sp + num2 / (den2 + 1e-10f)) * 0.5f;
  out_scalars[0] = sp;
  out_scalars[1] = tran;
}

extern "C" void kernel_launch(void* const* d_in, const int* in_sizes, int n_in,
                              void* d_out, int out_size, void* d_ws, size_t ws_size,
                              hipStream_t stream) {
  (void)in_sizes; (void)n_in; (void)out_size; (void)ws_size;
  const float* x    = (const float*)d_in[0];
  const float* E    = (const float*)d_in[1];
  const float* poly = (const float*)d_in[2];
  const float* mask = (const float*)d_in[3];
  const float* wp   = (const float*)d_in[4];
  const float* bp   = (const float*)d_in[5];
  float* out = (float*)d_out;
  float* ws  = (float*)d_ws;

  float* T    = ws + OFF_T;
  float* TscT = ws + OFF_TSC;
  float* CL   = ws + OFF_CL;
  float* EH   = ws + OFF_EH;
  float* Z    = ws + OFF_Z;
  unsigned int* ctr = (unsigned int*)(ws + OFF_CTR);

  k_build   <<<N_ / 256, 256, 0, stream>>>(E, poly, T, TscT, ctr);
  k_partial <<<G_ * B_, 640, 0, stream>>>(x, mask, T, CL, ctr);
  k_reduce  <<<(B_ * TWP_ * C_) / 256, 256, 0, stream>>>(CL, EH, ctr);
  k_zcombine<<<(B_ * (N_ / 16)) / 8, 256, 0, stream>>>(x, mask, TscT, EH, Z);
  k_out     <<<N_, 256, 0, stream>>>(E, wp, bp, Z, out);
  k_final   <<<1, 1, 0, stream>>>(ctr, out + (size_t)B_ * N_ * C_);
}